// LearnableSpectralHamiltonian_65987877535808
// MI455X (gfx1250) — compile-verified
//
#include <hip/hip_runtime.h>
#include <hip/hip_bf16.h>

// ---------------------------------------------------------------------------
// LearnableSpectralHamiltonian on gfx1250:
//   y = ifft2( fft2(x, ortho) * sqrt_lam, ortho ).real
// expressed as 12 real 64x64x64 matmuls per image on V_WMMA_F32_16X16X4_F32.
//
//   Fo = C - i*D,  C[u,h] = cos(2*pi*u*h/64)/8,  D[u,h] = sin(2*pi*u*h/64)/8
//   P  = C*x ; Q = D*x                      (U = P - iQ = Fo*x)
//   R1 = P*C - Q*D ; R2 = P*D + Q*C         (X = R1 - i*R2 = Fo*x*Fo)
//   W1 = S.*R1 ; W2 = S.*R2
//   Vr = C*W1 + D*W2 ; Vi = D*W1 - C*W2     (V = Fo^H * W)
//   Y  = Vr*C - Vi*D                        (Y = Re(V * Fo^H))
// ---------------------------------------------------------------------------

#define IMG_N     4096          // 64*64 elements per image
#define ST        68            // padded LDS row stride (floats): 272B = 16B-aligned
                                // rows, and column reads hit banks (4m+kb)%64 ->
                                // conflict-free b64 A-fragment loads
#define BUF       (64 * ST)     // one padded 64x64 LDS matrix

typedef __attribute__((ext_vector_type(2))) float v2f;
typedef __attribute__((ext_vector_type(8))) float v8f;
typedef int v4i __attribute__((__vector_size__(16)));

#define GLOBAL_AS __attribute__((address_space(1)))
#define LDS_AS    __attribute__((address_space(3)))

#if defined(__has_builtin)
#if __has_builtin(__builtin_amdgcn_global_load_async_to_lds_b128) && \
    __has_builtin(__builtin_amdgcn_s_wait_asynccnt)
#define USE_ASYNC_LDS 1
#endif
#endif

// ---------------------------------------------------------------------------
// Setup kernel (1 block, 256 threads): reproduce _sqrt_lambda exactly and
// tabulate the ortho DFT cos/sin matrices.
//   ws[0..4095]      = sqrt_lam (row-major 64x64)
//   ws[4096..8191]   = C  = cos(2*pi*u*h/64)/8
//   ws[8192..12287]  = D  = sin(2*pi*u*h/64)/8
// ---------------------------------------------------------------------------
__global__ void spectral_setup_kernel(const float* __restrict__ g1p,
                                      const float* __restrict__ raw_deltas,
                                      const int*   __restrict__ bin_idx,
                                      float*       __restrict__ ws) {
    __shared__ float red[256];
    __shared__ float means[2];
    const int tid = threadIdx.x;

    // monotone bin values: g[b] = g1 + cumsum(softplus(raw_deltas))
    float g[16];
    g[0] = g1p[0];
#pragma unroll
    for (int b = 1; b < 16; ++b) {
        float rd = raw_deltas[b - 1];
        float sp = fmaxf(rd, 0.0f) + log1pf(expf(-fabsf(rd)));   // stable softplus
        g[b] = g[b - 1] + sp;
    }

    // gather ell per pixel (16 pixels / thread), reduce mean
    float ell[16];
    float s = 0.0f;
#pragma unroll
    for (int k = 0; k < 16; ++k) {
        int p = tid + k * 256;
        ell[k] = g[bin_idx[p]];
        s += ell[k];
    }
    red[tid] = s;
    __syncthreads();
    for (int off = 128; off > 0; off >>= 1) {
        if (tid < off) red[tid] += red[tid + off];
        __syncthreads();
    }
    if (tid == 0) means[0] = red[0] * (1.0f / 4096.0f);
    __syncthreads();
    const float m1 = means[0];

    // center, clip, re-mean (of log lam == clipped ell)
    s = 0.0f;
#pragma unroll
    for (int k = 0; k < 16; ++k) {
        float e = ell[k] - m1;
        e = fminf(fmaxf(e, -3.0f), 3.0f);
        ell[k] = e;
        s += e;
    }
    __syncthreads();                 // red[] reuse
    red[tid] = s;
    __syncthreads();
    for (int off = 128; off > 0; off >>= 1) {
        if (tid < off) red[tid] += red[tid + off];
        __syncthreads();
    }
    if (tid == 0) means[1] = red[0] * (1.0f / 4096.0f);
    __syncthreads();
    const float m2    = means[1];
    const float denom = fmaxf(expf(m2), 1e-12f);

#pragma unroll
    for (int k = 0; k < 16; ++k) {
        int p = tid + k * 256;
        float lam = expf(ell[k]) / denom;
        ws[p] = sqrtf(lam);
        int u = p >> 6, h = p & 63;
        // e^{-2pi i u h / 64} periodic in (u*h) mod 64 -> exact angle
        float ang = (float)((u * h) & 63) * 0.09817477042468103f;  // 2*pi/64
        ws[4096 + p] = cosf(ang) * 0.125f;
        ws[8192 + p] = sinf(ang) * 0.125f;
    }
}

#ifdef USE_ASYNC_LDS
// One 16-byte async global->LDS chunk (tracked by ASYNCcnt).
// Signature per hipcc diagnostic: param0 = v4i addrspace(1)* (printed
// "__device__"), param1 = LDS destination, then imm offset + cpol.
__device__ __forceinline__ void async_copy16(const float* g, float* l) {
    __builtin_amdgcn_global_load_async_to_lds_b128(
        (GLOBAL_AS v4i*)(g), (LDS_AS v4i*)(l), 0, 0);
}
#endif

// ---------------------------------------------------------------------------
// Wave-level 64x64x64 f32 GEMM on V_WMMA_F32_16X16X4_F32.
// 8 wave32 per block; wave w owns the 16x32 strip (ti = w>>1, tj = {2(w&1),
// 2(w&1)+1}). One A fragment feeds two B tiles -> half the A-side LDS reads,
// and the two accumulators make consecutive WMMAs independent (no serial
// RAW accumulation chain).
// mode: 0 = assign, 1 = add into dst, 2 = subtract from dst.
//
// ISA fragment layouts (wave32):
//   A 16x4 : lanes 0-15 M=0..15 K={0,1}; lanes 16-31 M=0..15 K={2,3}  -> v2f
//   B 4x16 : lanes 0-15 rows {0,1};       lanes 16-31 rows {2,3}      -> v2f
//   C/D    : vgpr j, lanes 0-15 M=j, lanes 16-31 M=j+8, N=lane&15     -> v8f
// ---------------------------------------------------------------------------
__device__ __forceinline__ void wave_gemm(float* __restrict__ dst,
                                          const float* __restrict__ A,
                                          const float* __restrict__ B,
                                          int lane, int wave, int mode) {
    const int half = lane >> 4;        // 0 or 1
    const int l15  = lane & 15;
    const int ti   = wave >> 1;
    const int n0   = ((wave & 1) * 2) * 16 + l15;
    const int m    = ti * 16 + l15;
    v8f c0 = {};
    v8f c1 = {};
#pragma unroll 4
    for (int kk = 0; kk < 16; ++kk) {
        const int kb = kk * 4 + half * 2;
        v2f a = *(const v2f*)(A + m * ST + kb);   // ds_load_b64, conflict-free
        v2f b0, b1;
        b0.x = B[(kb + 0) * ST + n0];
        b0.y = B[(kb + 1) * ST + n0];
        b1.x = B[(kb + 0) * ST + n0 + 16];
        b1.y = B[(kb + 1) * ST + n0 + 16];
        c0 = __builtin_amdgcn_wmma_f32_16x16x4_f32(
                 false, a, false, b0, (short)0, c0, false, false);
        c1 = __builtin_amdgcn_wmma_f32_16x16x4_f32(
                 false, a, false, b1, (short)0, c1, false, false);
    }
#pragma unroll
    for (int j = 0; j < 8; ++j) {
        const int row = ti * 16 + half * 8 + j;
        float* p0 = dst + row * ST + n0;
        float* p1 = p0 + 16;
        if (mode == 0)      { *p0  = c0[j]; *p1  = c1[j]; }
        else if (mode == 1) { *p0 += c0[j]; *p1 += c1[j]; }
        else                { *p0 -= c0[j]; *p1 -= c1[j]; }
    }
}

// ---------------------------------------------------------------------------
// Main kernel: one block (256 thr = 8 wave32) per (n, d_f) image.
// Everything resident in LDS: C, D, x/Y, P/Vr, Q/Vi, R1, R2 (padded) + S.
// Staging uses GLOBAL_LOAD_ASYNC_TO_LDS_B128 when the toolchain declares it.
// ---------------------------------------------------------------------------
__global__ void __launch_bounds__(256)
spectral_apply_kernel(const float* __restrict__ x,
                      const float* __restrict__ ws,
                      float*       __restrict__ out) {
    extern __shared__ float smem[];
    float* Cc = smem;            // cos matrix
    float* Sn = Cc + BUF;        // sin matrix
    float* Xb = Sn + BUF;        // x, later Y
    float* P  = Xb + BUF;        // C*x, later Vr
    float* Q  = P  + BUF;        // D*x, later Vi
    float* R1 = Q  + BUF;        // Re(X), later W1
    float* R2 = R1 + BUF;        // -Im(X), later W2
    float* Sb = R2 + BUF;        // sqrt_lam, contiguous 4096

    const int tid  = threadIdx.x;
    const int lane = tid & 31;
    const int wave = tid >> 5;
    const size_t img = blockIdx.x;
    const float* xin = x + img * (size_t)IMG_N;

#ifdef USE_ASYNC_LDS
    // Stage x, C, D, S directly into LDS (16B chunks, ASYNCcnt-tracked).
#pragma unroll
    for (int k = 0; k < 4; ++k) {
        int q  = tid + k * 256;          // 16-byte chunk id, 1024 per matrix
        int r  = q >> 4;
        int c4 = (q & 15) * 4;
        async_copy16(xin + r * 64 + c4,        Xb + r * ST + c4);
        async_copy16(ws + 4096 + r * 64 + c4,  Cc + r * ST + c4);
        async_copy16(ws + 8192 + r * 64 + c4,  Sn + r * ST + c4);
        async_copy16(ws + q * 4,               Sb + q * 4);
    }
    __builtin_amdgcn_s_wait_asynccnt(0);
    __syncthreads();
#else
#pragma unroll
    for (int k = 0; k < 16; ++k) {
        int i = tid + k * 256;
        int r = i >> 6, cN = i & 63;
        Cc[r * ST + cN] = ws[4096 + i];
        Sn[r * ST + cN] = ws[8192 + i];
        Xb[r * ST + cN] = xin[i];
        Sb[i]           = ws[i];
    }
    __syncthreads();
#endif

    // forward, rows: U = Fo * x  (P = C x, Q = D x)
    wave_gemm(P, Cc, Xb, lane, wave, 0);
    wave_gemm(Q, Sn, Xb, lane, wave, 0);
    __syncthreads();

    // forward, cols: X = U * Fo  (R1 = PC - QD, R2 = PD + QC)
    wave_gemm(R1, P, Cc, lane, wave, 0);
    wave_gemm(R1, Q, Sn, lane, wave, 2);
    wave_gemm(R2, P, Sn, lane, wave, 0);
    wave_gemm(R2, Q, Cc, lane, wave, 1);
    __syncthreads();

    // spectral scale: W = S .* X
#pragma unroll
    for (int k = 0; k < 16; ++k) {
        int i = tid + k * 256;
        int r = i >> 6, cN = i & 63;
        float sv = Sb[i];
        R1[r * ST + cN] *= sv;
        R2[r * ST + cN] *= sv;
    }
    __syncthreads();

    // inverse, rows: V = Fo^H * W  (Vr = C W1 + D W2, Vi = D W1 - C W2)
    wave_gemm(P, Cc, R1, lane, wave, 0);
    wave_gemm(P, Sn, R2, lane, wave, 1);
    wave_gemm(Q, Sn, R1, lane, wave, 0);
    wave_gemm(Q, Cc, R2, lane, wave, 2);
    __syncthreads();

    // inverse, cols, real part only: Y = Vr C - Vi D
    wave_gemm(Xb, P, Cc, lane, wave, 0);
    wave_gemm(Xb, Q, Sn, lane, wave, 2);
    __syncthreads();

    float* yout = out + img * (size_t)IMG_N;
#pragma unroll
    for (int k = 0; k < 16; ++k) {
        int i = tid + k * 256;
        int r = i >> 6, cN = i & 63;
        yout[i] = Xb[r * ST + cN];
    }
}

// ---------------------------------------------------------------------------
extern "C" void kernel_launch(void* const* d_in, const int* in_sizes, int n_in,
                              void* d_out, int out_size, void* d_ws, size_t ws_size,
                              hipStream_t stream) {
    const float* x          = (const float*)d_in[0];   // (16,768,64,64) f32
    const float* g1         = (const float*)d_in[1];   // scalar
    const float* raw_deltas = (const float*)d_in[2];   // (15,)
    const int*   bin_idx    = (const int*)d_in[3];     // (64,64) int32
    float*       ws         = (float*)d_ws;            // S | C | D  (3*4096 f32)
    float*       out        = (float*)d_out;

    const int n_img = in_sizes[0] / IMG_N;             // 12288

    spectral_setup_kernel<<<1, 256, 0, stream>>>(g1, raw_deltas, bin_idx, ws);

    const size_t smem_bytes = (size_t)(7 * BUF + IMG_N) * sizeof(float); // ~135 KB
    spectral_apply_kernel<<<n_img, 256, smem_bytes, stream>>>(x, ws, out);
}